// TempVAE_34033320853995
// MI455X (gfx1250) — compile-verified
//
#include <hip/hip_runtime.h>
#include <cmath>

typedef float v2f __attribute__((ext_vector_type(2)));
typedef float v8f __attribute__((ext_vector_type(8)));

#define Bb    2048
#define Tt    1024
#define RDIM  64
#define LDIM  10
#define HDIM  16
#define H3    48
#define kLOG2PI 1.8378770664093453f

// ---------------- parameter bundle (setup_inputs dict order) ----------------
struct P {
  const float *R, *eps;
  const float *f_wi, *f_bi, *f_wh, *f_bh;        // inf_fwd
  const float *b_wi, *b_bi, *b_wh, *b_bh;        // inf_bwd
  const float *z_wi, *z_bi, *z_wh, *z_bh;        // inf_z
  const float *imu_w1,*imu_b1,*imu_w2,*imu_b2,*imu_w3,*imu_b3;   // inf_mu
  const float *isg_w1,*isg_b1,*isg_w2,*isg_b2,*isg_w3,*isg_b3;   // inf_sig
  const float *gz_wi,*gz_bi,*gz_wh,*gz_bh;       // gen_z
  const float *gr_wi,*gr_bi,*gr_wh,*gr_bh;       // gen_r
  const float *pmu_w1,*pmu_b1,*pmu_w2,*pmu_b2,*pmu_w3,*pmu_b3;   // gen_mu_z
  const float *psg_w1,*psg_b1,*psg_w2,*psg_b2,*psg_w3,*psg_b3;   // gen_sig_z
  const float *rmu_w1,*rmu_b1,*rmu_w2,*rmu_b2,*rmu_w3,*rmu_b3;   // gen_mu_r
  const float *cov_w,*cov_b;
};

// ---------------- WMMA f32 16x16x4 ----------------
static __device__ __forceinline__ v8f wmma4(v2f a, v2f b, v8f c) {
  return __builtin_amdgcn_wmma_f32_16x16x4_f32(false, a, false, b, (short)0, c,
                                               false, false);
}
static __device__ __forceinline__ v8f vzero() {
  v8f v;
#pragma unroll
  for (int i = 0; i < 8; ++i) v[i] = 0.f;
  return v;
}
static __device__ __forceinline__ void vz3(v8f a[3]) {
#pragma unroll
  for (int n = 0; n < 3; ++n) a[n] = vzero();
}
static __device__ __forceinline__ float sigm_(float x) { return 1.f / (1.f + __expf(-x)); }
static __device__ __forceinline__ float tanh_(float x) { return 1.f - 2.f / (__expf(2.f * x) + 1.f); }

// A fragment from row-major LDS tile: row m = lane&15, cols k0,k0+1
static __device__ __forceinline__ v2f afragL(const float* t, int s, int nn, int k0) {
  v2f a; a[0] = t[nn * s + k0]; a[1] = t[nn * s + k0 + 1]; return a;
}
// B fragment from LDS B-image [KP][stride]: rows k0,k0+1 at column n
static __device__ __forceinline__ v2f bfragL(const float* bm, int s, int k0, int n) {
  v2f b; b[0] = bm[k0 * s + n]; b[1] = bm[(k0 + 1) * s + n]; return b;
}
// D-fragment scatter to row-major LDS tile at column offset
static __device__ __forceinline__ void storeD(float* t, int s, int co, v8f d, int grp, int nn) {
#pragma unroll
  for (int i = 0; i < 8; ++i) t[(i + 8 * grp) * s + co + nn] = d[i];
}

// N=48 GEMM, A from LDS tile, B from LDS image (stride 48)
static __device__ __forceinline__ void gemm48_lds(v8f acc[3], const float* At, int sA,
                                                  const float* Bm, int KT, int grp, int nn) {
  for (int kt = 0; kt < KT; ++kt) {
    int k0 = kt * 4 + 2 * grp;
    v2f a = afragL(At, sA, nn, k0);
#pragma unroll
    for (int nt = 0; nt < 3; ++nt)
      acc[nt] = wmma4(a, bfragL(Bm, H3, k0, nt * 16 + nn), acc[nt]);
  }
}
// N=48 GEMM, A from LDS tile, B streamed from global torch-layout w[48][K]
static __device__ __forceinline__ void gemm48_gB(v8f acc[3], const float* At, int sA,
                                                 const float* wg, int K, int KT, int grp, int nn) {
  for (int kt = 0; kt < KT; ++kt) {
    int k0 = kt * 4 + 2 * grp;
    int k0c = (k0 + 1 < K) ? k0 : 0;   // pad rows: A side is zero there
    v2f a = afragL(At, sA, nn, k0);
#pragma unroll
    for (int nt = 0; nt < 3; ++nt) {
      const float* q = wg + (nt * 16 + nn) * K + k0c;
      v2f b; b[0] = q[0]; b[1] = q[1];
      acc[nt] = wmma4(a, b, acc[nt]);
    }
  }
}
// N=16 GEMM, A + B both in LDS (B stride 16), K=16
static __device__ __forceinline__ v8f gemm16(const float* At, int sA, const float* Bm,
                                             int grp, int nn) {
  v8f c = vzero();
#pragma unroll
  for (int kt = 0; kt < 4; ++kt) {
    int k0 = kt * 4 + 2 * grp;
    c = wmma4(afragL(At, sA, nn, k0), bfragL(Bm, 16, k0, nn), c);
  }
  return c;
}

// GRU gate combine (torch order r,z,n); hD updated in place
static __device__ __forceinline__ void gru_combine(const v8f xp[3], const v8f hp[3],
                                                   const float* bi, const float* bh,
                                                   float hD[8], int nn) {
  float bir[3], bhr[3];
#pragma unroll
  for (int nt = 0; nt < 3; ++nt) { bir[nt] = bi[nt * 16 + nn]; bhr[nt] = bh[nt * 16 + nn]; }
#pragma unroll
  for (int i = 0; i < 8; ++i) {
    float xr = xp[0][i] + bir[0], xz = xp[1][i] + bir[1], xn = xp[2][i] + bir[2];
    float hr = hp[0][i] + bhr[0], hz = hp[1][i] + bhr[1], hn = hp[2][i] + bhr[2];
    float r = sigm_(xr + hr);
    float z = sigm_(xz + hz);
    float n = tanh_(xn + r * hn);
    hD[i] = (1.f - z) * n + z * hD[i];
  }
}

// ---------------- cooperative LDS weight loaders ----------------
static __device__ void loadB_(float* dst, const float* w, int K, int N, int KP, int NP,
                              int tid, int nth) {
  for (int i = tid; i < KP * NP; i += nth) {
    int k = i / NP, n = i % NP;
    dst[i] = (k < K && n < N) ? w[n * K + k] : 0.f;
  }
}
static __device__ void loadV_(float* dst, const float* v, int N, int NP, int tid, int nth) {
  for (int i = tid; i < NP; i += nth) dst[i] = (i < N) ? v[i] : 0.f;
}

struct MLPW { float Bw1[256]; float b1[16]; float Bw2[256]; float b2[16];
              float Bw3[256]; float b3[16]; };

static __device__ void load_mlp(MLPW* M, const float* w1, const float* b1,
                                const float* w2, const float* b2,
                                const float* w3, const float* b3, int N3,
                                int tid, int nth) {
  loadB_(M->Bw1, w1, 16, 16, 16, 16, tid, nth);
  loadV_(M->b1, b1, 16, 16, tid, nth);
  loadB_(M->Bw2, w2, 16, 16, 16, 16, tid, nth);
  loadV_(M->b2, b2, 16, 16, tid, nth);
  loadB_(M->Bw3, w3, 16, N3, 16, 16, tid, nth);   // cols >= N3 zero
  loadV_(M->b3, b3, N3, 16, tid, nth);
}

// two hidden relu layers; result left in scr (16x16)
static __device__ __forceinline__ void mlp_hidden(const float* Ain, float* scr,
                                                  const MLPW* W, int grp, int nn) {
  v8f h = gemm16(Ain, 16, W->Bw1, grp, nn);
  float b = W->b1[nn];
#pragma unroll
  for (int i = 0; i < 8; ++i) h[i] = fmaxf(h[i] + b, 0.f);
  storeD(scr, 16, 0, h, grp, nn);
  h = gemm16(scr, 16, W->Bw2, grp, nn);
  b = W->b2[nn];
#pragma unroll
  for (int i = 0; i < 8; ++i) h[i] = fmaxf(h[i] + b, 0.f);
  storeD(scr, 16, 0, h, grp, nn);
}
static __device__ __forceinline__ v8f mlp_out(const float* scr, const float* Bw3,
                                              const float* b3, int grp, int nn) {
  v8f o = gemm16(scr, 16, Bw3, grp, nn);
  float b = b3[nn];
#pragma unroll
  for (int i = 0; i < 8; ++i) o[i] += b;
  return o;
}

// ================= kernel 1: backward GRU, h_b -> workspace =================
__global__ __launch_bounds__(128) void vae_bwd_k(P p, float* __restrict__ hb,
                                                 float* __restrict__ acc) {
  __shared__ float sBwh[16 * H3];
  __shared__ float sbi[H3], sbh[H3];
  __shared__ float sH[4][16 * 16];
  const int tid = threadIdx.x, nth = blockDim.x;
  loadB_(sBwh, p.b_wh, 16, H3, 16, H3, tid, nth);
  loadV_(sbi, p.b_bi, H3, H3, tid, nth);
  loadV_(sbh, p.b_bh, H3, H3, tid, nth);
  for (int i = tid; i < 4 * 256; i += nth) (&sH[0][0])[i] = 0.f;
  if (blockIdx.x == 0 && tid == 0) { acc[0] = 0.f; acc[1] = 0.f; }
  __syncthreads();

  const int lane = tid & 31, wv = tid >> 5;
  const int grp = lane >> 4, nn = lane & 15;
  const int b0 = (blockIdx.x * 4 + wv) * 16;
  float* hst = &sH[wv][0];
  float hD[8];
#pragma unroll
  for (int i = 0; i < 8; ++i) hD[i] = 0.f;

  for (int t = Tt - 1; t >= 0; --t) {
    const float* Rrow = p.R + ((size_t)(b0 + nn) * Tt + t) * RDIM;
    if (t > 0) __builtin_prefetch(Rrow - RDIM, 0, 0);
    v8f xp[3]; vz3(xp);
    for (int kt = 0; kt < 16; ++kt) {
      int k0 = kt * 4 + 2 * grp;
      v2f a; a[0] = Rrow[k0]; a[1] = Rrow[k0 + 1];
#pragma unroll
      for (int nt = 0; nt < 3; ++nt) {
        const float* q = p.b_wi + (nt * 16 + nn) * RDIM + k0;
        v2f b; b[0] = q[0]; b[1] = q[1];
        xp[nt] = wmma4(a, b, xp[nt]);
      }
    }
    v8f hp[3]; vz3(hp);
    gemm48_lds(hp, hst, 16, sBwh, 4, grp, nn);
    gru_combine(xp, hp, sbi, sbh, hD, nn);
#pragma unroll
    for (int i = 0; i < 8; ++i) {
      int m = i + 8 * grp;
      hst[m * 16 + nn] = hD[i];
      hb[((size_t)(b0 + m) * Tt + t) * HDIM + nn] = hD[i];
    }
  }
}

// ====== kernel 2: fused fwd GRU + inference + generative + loss accum =======
__global__ __launch_bounds__(64) void vae_fused_k(P p, const float* __restrict__ hb,
                                                  float* __restrict__ acc) {
  __shared__ float s_bi_f[H3], s_bh_f[H3], s_Bwh_f[16 * H3];
  __shared__ float s_bi_z[H3], s_bh_z[H3], s_Bwh_z[16 * H3];
  __shared__ MLPW  s_mlp[5];  // inf_mu, inf_sig, gen_mu_z, gen_sig_z, gen_mu_r(l1,l2)
  __shared__ float s_Bwi_gz[12 * H3], s_bi_gz[H3], s_Bwh_gz[16 * H3], s_bh_gz[H3];
  __shared__ float s_Bwi_gr[12 * H3], s_bi_gr[H3], s_Bwh_gr[16 * H3], s_bh_gr[H3];
  __shared__ float s_Bw3_mr[16 * 64], s_b3_mr[64];
  __shared__ float s_Bcov[16 * 64],   s_bcov[64];
  __shared__ float s_xz[2][16 * H3];   // [z_prev | h_f | h_b | pad]
  __shared__ float s_zt[2][16 * 12];
  __shared__ float s_hq[2][256], s_hz[2][256], s_hr[2][256], s_scr[2][256];

  const int tid = threadIdx.x, nth = blockDim.x;
  loadV_(s_bi_f, p.f_bi, H3, H3, tid, nth);
  loadV_(s_bh_f, p.f_bh, H3, H3, tid, nth);
  loadB_(s_Bwh_f, p.f_wh, 16, H3, 16, H3, tid, nth);
  loadV_(s_bi_z, p.z_bi, H3, H3, tid, nth);
  loadV_(s_bh_z, p.z_bh, H3, H3, tid, nth);
  loadB_(s_Bwh_z, p.z_wh, 16, H3, 16, H3, tid, nth);
  load_mlp(&s_mlp[0], p.imu_w1, p.imu_b1, p.imu_w2, p.imu_b2, p.imu_w3, p.imu_b3, LDIM, tid, nth);
  load_mlp(&s_mlp[1], p.isg_w1, p.isg_b1, p.isg_w2, p.isg_b2, p.isg_w3, p.isg_b3, LDIM, tid, nth);
  load_mlp(&s_mlp[2], p.pmu_w1, p.pmu_b1, p.pmu_w2, p.pmu_b2, p.pmu_w3, p.pmu_b3, LDIM, tid, nth);
  load_mlp(&s_mlp[3], p.psg_w1, p.psg_b1, p.psg_w2, p.psg_b2, p.psg_w3, p.psg_b3, LDIM, tid, nth);
  load_mlp(&s_mlp[4], p.rmu_w1, p.rmu_b1, p.rmu_w2, p.rmu_b2, p.rmu_w1, p.rmu_b1, 0, tid, nth);
  loadB_(s_Bwi_gz, p.gz_wi, LDIM, H3, 12, H3, tid, nth);
  loadV_(s_bi_gz, p.gz_bi, H3, H3, tid, nth);
  loadB_(s_Bwh_gz, p.gz_wh, 16, H3, 16, H3, tid, nth);
  loadV_(s_bh_gz, p.gz_bh, H3, H3, tid, nth);
  loadB_(s_Bwi_gr, p.gr_wi, LDIM, H3, 12, H3, tid, nth);
  loadV_(s_bi_gr, p.gr_bi, H3, H3, tid, nth);
  loadB_(s_Bwh_gr, p.gr_wh, 16, H3, 16, H3, tid, nth);
  loadV_(s_bh_gr, p.gr_bh, H3, H3, tid, nth);
  loadB_(s_Bw3_mr, p.rmu_w3, 16, 64, 16, 64, tid, nth);
  loadV_(s_b3_mr, p.rmu_b3, 64, 64, tid, nth);
  loadB_(s_Bcov, p.cov_w, 16, 64, 16, 64, tid, nth);
  loadV_(s_bcov, p.cov_b, 64, 64, tid, nth);
  for (int i = tid; i < 2 * 16 * H3; i += nth) (&s_xz[0][0])[i] = 0.f;
  for (int i = tid; i < 2 * 16 * 12; i += nth) (&s_zt[0][0])[i] = 0.f;
  for (int i = tid; i < 2 * 256; i += nth) {
    (&s_hq[0][0])[i] = 0.f; (&s_hz[0][0])[i] = 0.f;
    (&s_hr[0][0])[i] = 0.f; (&s_scr[0][0])[i] = 0.f;
  }
  __syncthreads();

  const int lane = tid & 31, wv = tid >> 5;
  const int grp = lane >> 4, nn = lane & 15;
  const int b0 = (blockIdx.x * 2 + wv) * 16;
  float* xz = &s_xz[wv][0];
  float* zt = &s_zt[wv][0];
  float* hq_t = &s_hq[wv][0];
  float* hz_t = &s_hz[wv][0];
  float* hr_t = &s_hr[wv][0];
  float* scr = &s_scr[wv][0];

  float hf[8], hqr[8], hzr[8], hrr[8];
#pragma unroll
  for (int i = 0; i < 8; ++i) { hf[i] = hqr[i] = hzr[i] = hrr[i] = 0.f; }
  float acc_rec = 0.f, acc_kl = 0.f;
  const float msk = (nn < LDIM) ? 1.f : 0.f;

  for (int t = 0; t < Tt; ++t) {
    const float* Rrow = p.R + ((size_t)(b0 + nn) * Tt + t) * RDIM;
    if (t + 1 < Tt) __builtin_prefetch(Rrow + RDIM, 0, 0);

    // (1) forward GRU: xp = R @ wi_f^T, hp = h_f @ wh_f^T
    v8f xp[3]; vz3(xp);
    for (int kt = 0; kt < 16; ++kt) {
      int k0 = kt * 4 + 2 * grp;
      v2f a; a[0] = Rrow[k0]; a[1] = Rrow[k0 + 1];
#pragma unroll
      for (int nt = 0; nt < 3; ++nt) {
        const float* q = p.f_wi + (nt * 16 + nn) * RDIM + k0;
        v2f b; b[0] = q[0]; b[1] = q[1];
        xp[nt] = wmma4(a, b, xp[nt]);
      }
    }
    v8f hp[3]; vz3(hp);
    gemm48_lds(hp, xz + 10, H3, s_Bwh_f, 4, grp, nn);
    gru_combine(xp, hp, s_bi_f, s_bh_f, hf, nn);
    {
      v8f d;
#pragma unroll
      for (int i = 0; i < 8; ++i) d[i] = hf[i];
      storeD(xz, H3, 10, d, grp, nn);             // h_f -> cols 10..25
    }

    // (2) stream in h_b -> cols 26..41
#pragma unroll
    for (int i = 0; i < 8; ++i) {
      int m = i + 8 * grp;
      xz[m * H3 + 26 + nn] = hb[((size_t)(b0 + m) * Tt + t) * HDIM + nn];
    }

    // (3) inf_z GRU over [z_prev | h_f | h_b]
    vz3(xp);
    gemm48_gB(xp, xz, H3, p.z_wi, 42, 11, grp, nn);
    vz3(hp);
    gemm48_lds(hp, hq_t, 16, s_Bwh_z, 4, grp, nn);
    gru_combine(xp, hp, s_bi_z, s_bh_z, hqr, nn);
    {
      v8f d;
#pragma unroll
      for (int i = 0; i < 8; ++i) d[i] = hqr[i];
      storeD(hq_t, 16, 0, d, grp, nn);
    }

    // (4) q MLPs
    mlp_hidden(hq_t, scr, &s_mlp[0], grp, nn);
    v8f qmu = mlp_out(scr, s_mlp[0].Bw3, s_mlp[0].b3, grp, nn);
    mlp_hidden(hq_t, scr, &s_mlp[1], grp, nn);
    v8f qsr = mlp_out(scr, s_mlp[1].Bw3, s_mlp[1].b3, grp, nn);

    // (5) gen_z GRU (consumes z_prev at cols 0..9, before overwrite)
    vz3(xp);
    gemm48_lds(xp, xz, H3, s_Bwi_gz, 3, grp, nn);
    vz3(hp);
    gemm48_lds(hp, hz_t, 16, s_Bwh_gz, 4, grp, nn);
    gru_combine(xp, hp, s_bi_gz, s_bh_gz, hzr, nn);
    {
      v8f d;
#pragma unroll
      for (int i = 0; i < 8; ++i) d[i] = hzr[i];
      storeD(hz_t, 16, 0, d, grp, nn);
    }

    // (6) rsample z = mu + sig*eps; publish z (cols 0..9 and zt tile)
    v8f qsig, zv;
#pragma unroll
    for (int i = 0; i < 8; ++i) {
      int m = i + 8 * grp;
      int nc = (nn < LDIM) ? nn : (LDIM - 1);
      float e = msk * p.eps[((size_t)(b0 + m) * Tt + t) * LDIM + nc];
      qsig[i] = __expf(qsr[i]);
      zv[i] = qmu[i] + qsig[i] * e;
    }
    if (nn < LDIM) {
#pragma unroll
      for (int i = 0; i < 8; ++i) {
        int m = i + 8 * grp;
        xz[m * H3 + nn] = zv[i];
        zt[m * 12 + nn] = zv[i];
      }
    }

    // (7) gen_r GRU over z_t
    vz3(xp);
    gemm48_lds(xp, zt, 12, s_Bwi_gr, 3, grp, nn);
    vz3(hp);
    gemm48_lds(hp, hr_t, 16, s_Bwh_gr, 4, grp, nn);
    gru_combine(xp, hp, s_bi_gr, s_bh_gr, hrr, nn);
    {
      v8f d;
#pragma unroll
      for (int i = 0; i < 8; ++i) d[i] = hrr[i];
      storeD(hr_t, 16, 0, d, grp, nn);
    }

    // (8) prior MLPs
    mlp_hidden(hz_t, scr, &s_mlp[2], grp, nn);
    v8f pmu = mlp_out(scr, s_mlp[2].Bw3, s_mlp[2].b3, grp, nn);
    mlp_hidden(hz_t, scr, &s_mlp[3], grp, nn);
    v8f psr = mlp_out(scr, s_mlp[3].Bw3, s_mlp[3].b3, grp, nn);

    // (9) likelihood: mu_r (3-layer) + diag cov, fused per 16-col slab
    mlp_hidden(hr_t, scr, &s_mlp[4], grp, nn);  // h2 of gen_mu_r in scr
    for (int n4 = 0; n4 < 4; ++n4) {
      v8f cm = vzero(), cd = vzero();
#pragma unroll
      for (int kt = 0; kt < 4; ++kt) {
        int k0 = kt * 4 + 2 * grp;
        v2f am = afragL(scr, 16, nn, k0);
        v2f bm; bm[0] = s_Bw3_mr[k0 * 64 + n4 * 16 + nn];
        bm[1] = s_Bw3_mr[(k0 + 1) * 64 + n4 * 16 + nn];
        cm = wmma4(am, bm, cm);
        v2f ah = afragL(hr_t, 16, nn, k0);
        v2f bc; bc[0] = s_Bcov[k0 * 64 + n4 * 16 + nn];
        bc[1] = s_Bcov[(k0 + 1) * 64 + n4 * 16 + nn];
        cd = wmma4(ah, bc, cd);
      }
      float bmu = s_b3_mr[n4 * 16 + nn], bcv = s_bcov[n4 * 16 + nn];
#pragma unroll
      for (int i = 0; i < 8; ++i) {
        int m = i + 8 * grp;
        float var = __expf(cd[i] + bcv) + 1e-6f;
        float rv = p.R[((size_t)(b0 + m) * Tt + t) * RDIM + n4 * 16 + nn];
        float dd = rv - (cm[i] + bmu);
        acc_rec += 0.5f * (dd * dd / var + __logf(var) + kLOG2PI);
      }
    }

    // (10) KL(q||p), only real latent columns
#pragma unroll
    for (int i = 0; i < 8; ++i) {
      float d = qmu[i] - pmu[i];
      float ps2 = __expf(2.f * psr[i]);
      acc_kl += msk * ((psr[i] - qsr[i]) +
                       (qsig[i] * qsig[i] + d * d) / (2.f * ps2) - 0.5f);
    }
  }

  // wave32 reduction, then one atomic per wave
  for (int off = 16; off > 0; off >>= 1) {
    acc_rec += __shfl_xor(acc_rec, off);
    acc_kl  += __shfl_xor(acc_kl, off);
  }
  if (lane == 0) {
    atomicAdd(&acc[0], acc_rec);
    atomicAdd(&acc[1], acc_kl);
  }
}

// ================= kernel 3: L2 Frobenius + ELBO assembly ==================
static __device__ float fro_(const float* w, int n) {
  float s = 0.f;
  for (int i = 0; i < n; ++i) s += w[i] * w[i];
  return sqrtf(s);
}
__global__ __launch_bounds__(32) void vae_fin_k(P p, const float* __restrict__ acc,
                                                float* __restrict__ out) {
  if (threadIdx.x != 0 || blockIdx.x != 0) return;
  float l2 = 0.f;
  l2 += fro_(p.imu_w1, 256) + fro_(p.imu_w2, 256) + fro_(p.imu_w3, 160);
  l2 += fro_(p.isg_w1, 256) + fro_(p.isg_w2, 256) + fro_(p.isg_w3, 160);
  l2 += fro_(p.pmu_w1, 256) + fro_(p.pmu_w2, 256) + fro_(p.pmu_w3, 160);
  l2 += fro_(p.psg_w1, 256) + fro_(p.psg_w2, 256) + fro_(p.psg_w3, 160);
  l2 += fro_(p.rmu_w1, 256) + fro_(p.rmu_w2, 256) + fro_(p.rmu_w3, 1024);
  l2 *= 0.01f;
  float recon = acc[0] / (float)Bb;
  float kl = acc[1] / (float)Bb;
  out[0] = -recon - kl - l2;
  out[1] = recon;
  out[2] = kl;
}

// ============================== launcher ===================================
extern "C" void kernel_launch(void* const* d_in, const int* in_sizes, int n_in,
                              void* d_out, int out_size, void* d_ws, size_t ws_size,
                              hipStream_t stream) {
  (void)in_sizes; (void)n_in; (void)out_size; (void)ws_size;
  P p;
  int i = 0;
  p.R = (const float*)d_in[i++]; p.eps = (const float*)d_in[i++];
  p.f_wi = (const float*)d_in[i++]; p.f_bi = (const float*)d_in[i++];
  p.f_wh = (const float*)d_in[i++]; p.f_bh = (const float*)d_in[i++];
  p.b_wi = (const float*)d_in[i++]; p.b_bi = (const float*)d_in[i++];
  p.b_wh = (const float*)d_in[i++]; p.b_bh = (const float*)d_in[i++];
  p.z_wi = (const float*)d_in[i++]; p.z_bi = (const float*)d_in[i++];
  p.z_wh = (const float*)d_in[i++]; p.z_bh = (const float*)d_in[i++];
  p.imu_w1 = (const float*)d_in[i++]; p.imu_b1 = (const float*)d_in[i++];
  p.imu_w2 = (const float*)d_in[i++]; p.imu_b2 = (const float*)d_in[i++];
  p.imu_w3 = (const float*)d_in[i++]; p.imu_b3 = (const float*)d_in[i++];
  p.isg_w1 = (const float*)d_in[i++]; p.isg_b1 = (const float*)d_in[i++];
  p.isg_w2 = (const float*)d_in[i++]; p.isg_b2 = (const float*)d_in[i++];
  p.isg_w3 = (const float*)d_in[i++]; p.isg_b3 = (const float*)d_in[i++];
  p.gz_wi = (const float*)d_in[i++]; p.gz_bi = (const float*)d_in[i++];
  p.gz_wh = (const float*)d_in[i++]; p.gz_bh = (const float*)d_in[i++];
  p.gr_wi = (const float*)d_in[i++]; p.gr_bi = (const float*)d_in[i++];
  p.gr_wh = (const float*)d_in[i++]; p.gr_bh = (const float*)d_in[i++];
  p.pmu_w1 = (const float*)d_in[i++]; p.pmu_b1 = (const float*)d_in[i++];
  p.pmu_w2 = (const float*)d_in[i++]; p.pmu_b2 = (const float*)d_in[i++];
  p.pmu_w3 = (const float*)d_in[i++]; p.pmu_b3 = (const float*)d_in[i++];
  p.psg_w1 = (const float*)d_in[i++]; p.psg_b1 = (const float*)d_in[i++];
  p.psg_w2 = (const float*)d_in[i++]; p.psg_b2 = (const float*)d_in[i++];
  p.psg_w3 = (const float*)d_in[i++]; p.psg_b3 = (const float*)d_in[i++];
  p.rmu_w1 = (const float*)d_in[i++]; p.rmu_b1 = (const float*)d_in[i++];
  p.rmu_w2 = (const float*)d_in[i++]; p.rmu_b2 = (const float*)d_in[i++];
  p.rmu_w3 = (const float*)d_in[i++]; p.rmu_b3 = (const float*)d_in[i++];
  p.cov_w = (const float*)d_in[i++]; p.cov_b = (const float*)d_in[i++];

  float* ws = (float*)d_ws;
  float* hb = ws;                                    // [B][T][H] = 128 MB
  float* acc = ws + (size_t)Bb * Tt * HDIM;          // 2 accumulators

  vae_bwd_k<<<Bb / (16 * 4), 128, 0, stream>>>(p, hb, acc);
  vae_fused_k<<<Bb / (16 * 2), 64, 0, stream>>>(p, hb, acc);
  vae_fin_k<<<1, 32, 0, stream>>>(p, acc, (float*)d_out);
}